// PeriodGuidedMultiScaleRouter_90675349553300
// MI455X (gfx1250) — compile-verified
//
#include <hip/hip_runtime.h>
#include <hip/hip_bf16.h>

typedef __attribute__((ext_vector_type(2))) float v2f;
typedef __attribute__((ext_vector_type(8))) float v8f;

#define SEQ_LEN      512
#define NUM_FEATURES 64
#define NUM_NODES    128
#define BATCH        32
#define NUM_FREQS    256
#define HIDDEN       1024
#define NUM_EXPERTS  43
#define TOPK         3

#define TWO_PI 6.28318530717958647692f
#define INV_SQRT_512 0.04419417382415922f

// ---------------------------------------------------------------------------
// Kernel 1: s[b,l] = mean_n( sum_c x[b,c,l,n]*fc_w[c] ) + fc_b
// x layout: [B][C][L][N], N contiguous. One block per (b,l); 256 threads
// stream 64*128 = 8192 floats = 2048 float4 (fully coalesced 128B requests).
// This kernel is the 512MB / 23.3TB/s memory-bound floor of the problem.
// ---------------------------------------------------------------------------
__global__ __launch_bounds__(256) void k_reduce(const float* __restrict__ x,
                                                const float* __restrict__ fc_w,
                                                const float* __restrict__ fc_b,
                                                float* __restrict__ s) {
  const int blk = blockIdx.x;
  const int b = blk >> 9;        // / SEQ_LEN
  const int l = blk & (SEQ_LEN - 1);
  const long base = (long)b * (NUM_FEATURES * SEQ_LEN * NUM_NODES) + (long)l * NUM_NODES;
  const int tid = threadIdx.x;

  float partial = 0.f;
#pragma unroll
  for (int it = 0; it < 8; ++it) {
    const int v = tid + it * 256;     // float4 index in [0, 2048)
    const int c = v >> 5;             // 32 float4 per c-row (128 floats)
    const int r = v & 31;
    const float4 d = *(const float4*)(x + base + (long)c * (SEQ_LEN * NUM_NODES) + r * 4);
    partial += fc_w[c] * (d.x + d.y + d.z + d.w);
  }
  // wave32 reduce
#pragma unroll
  for (int off = 16; off > 0; off >>= 1)
    partial += __shfl_down(partial, off, 32);
  __shared__ float red[8];
  if ((tid & 31) == 0) red[tid >> 5] = partial;
  __syncthreads();
  if (tid == 0) {
    float t = 0.f;
#pragma unroll
    for (int w = 0; w < 8; ++w) t += red[w];
    s[b * SEQ_LEN + l] = t * (1.0f / NUM_NODES) + fc_b[0];
  }
}

// ---------------------------------------------------------------------------
// Kernel 2: rfft(s, norm='ortho')[:,1:] via table-driven direct DFT.
// One block per batch row; 256 threads = freqs k=1..256. Twiddles in LDS.
// ---------------------------------------------------------------------------
__global__ __launch_bounds__(256) void k_rfft(const float* __restrict__ s,
                                              float* __restrict__ re,
                                              float* __restrict__ im) {
  __shared__ float sd[SEQ_LEN];
  __shared__ float ct[SEQ_LEN];
  __shared__ float st[SEQ_LEN];
  const int b = blockIdx.x;
  const int tid = threadIdx.x;
  for (int i = tid; i < SEQ_LEN; i += 256) {
    sd[i] = s[b * SEQ_LEN + i];
    const float ang = TWO_PI * (float)i / (float)SEQ_LEN;
    ct[i] = __cosf(ang);
    st[i] = __sinf(ang);
  }
  __syncthreads();
  const int k = tid + 1;            // freq 1..256 (DC dropped, Nyquist kept)
  float ar = 0.f, ai = 0.f;
  int idx = 0;
  for (int l = 0; l < SEQ_LEN; ++l) {
    const float v = sd[l];
    ar = fmaf(v, ct[idx], ar);
    ai = fmaf(v, st[idx], ai);
    idx = (idx + k) & (SEQ_LEN - 1);
  }
  re[b * NUM_FREQS + tid] = ar * INV_SQRT_512;
  im[b * NUM_FREQS + tid] = -ai * INV_SQRT_512;
}

// ---------------------------------------------------------------------------
// Kernel 3: complex layer 1 via V_WMMA_F32_16X16X4_F32.
// o1r = relu(re@w1[0] - im@w1[1] + b1[0]); o1i = relu(im@w1[0] + re@w1[1] + b1[1])
// Grid: (HIDDEN/16, BATCH/16) blocks, one wave each. 4 f32 accumulators
// (RA,RB,IA,IB) let us form both complex outputs without A/B negation.
// ---------------------------------------------------------------------------
__global__ __launch_bounds__(32) void k_layer1(const float* __restrict__ re,
                                               const float* __restrict__ im,
                                               const float* __restrict__ w1,
                                               const float* __restrict__ b1,
                                               float* __restrict__ o1r,
                                               float* __restrict__ o1i) {
  const int lane = threadIdx.x;
  const int n0 = blockIdx.x * 16;
  const int m0 = blockIdx.y * 16;
  const int m = lane & 15;
  const int half = lane >> 4;             // K half select for A/B layouts
  const int col = n0 + (lane & 15);
  const float* __restrict__ wa = w1;                         // w1[0]: [256][1024]
  const float* __restrict__ wb = w1 + NUM_FREQS * HIDDEN;    // w1[1]

  v8f accRA = {}, accRB = {}, accIA = {}, accIB = {};
  for (int k0 = 0; k0 < NUM_FREQS; k0 += 4) {
    const int ka = k0 + 2 * half;
    // A tiles (16x4 f32): lane = m + 16*(k/2), vgpr = k&1
    v2f aR, aI, bA, bB;
    const int arow = (m0 + m) * NUM_FREQS + ka;
    aR.x = re[arow];     aR.y = re[arow + 1];
    aI.x = im[arow];     aI.y = im[arow + 1];
    // B tiles (4x16 f32): vgpr v holds K=v (lanes 0-15), K=v+2 (lanes 16-31)
    bA.x = wa[(long)ka * HIDDEN + col];
    bA.y = wa[(long)(ka + 1) * HIDDEN + col];
    bB.x = wb[(long)ka * HIDDEN + col];
    bB.y = wb[(long)(ka + 1) * HIDDEN + col];

    accRA = __builtin_amdgcn_wmma_f32_16x16x4_f32(false, aR, false, bA, (short)0, accRA, false, false);
    accRB = __builtin_amdgcn_wmma_f32_16x16x4_f32(false, aR, false, bB, (short)0, accRB, false, false);
    accIA = __builtin_amdgcn_wmma_f32_16x16x4_f32(false, aI, false, bA, (short)0, accIA, false, false);
    accIB = __builtin_amdgcn_wmma_f32_16x16x4_f32(false, aI, false, bB, (short)0, accIB, false, false);
  }
  // C/D layout: vgpr v -> row = v + 8*half, col = lane&15
  const float br = b1[col];
  const float bi = b1[HIDDEN + col];
#pragma unroll
  for (int v = 0; v < 8; ++v) {
    const int row = m0 + v + 8 * half;
    const float orv = accRA[v] - accIB[v] + br;
    const float oiv = accIA[v] + accRB[v] + bi;
    o1r[row * HIDDEN + col] = fmaxf(orv, 0.f);
    o1i[row * HIDDEN + col] = fmaxf(oiv, 0.f);
  }
}

// ---------------------------------------------------------------------------
// Kernel 4: complex layer 2 + amplitude via V_WMMA_F32_16X16X4_F32.
// o2r = o1r@w2[0] - o1i@w2[1] + b2[0]; o2i = o1i@w2[0] + o1r@w2[1] + b2[1]
// amp = sqrt(o2r^2 + o2i^2).  Grid: (NUM_FREQS/16, BATCH/16), one wave each.
// ---------------------------------------------------------------------------
__global__ __launch_bounds__(32) void k_layer2(const float* __restrict__ o1r,
                                               const float* __restrict__ o1i,
                                               const float* __restrict__ w2,
                                               const float* __restrict__ b2,
                                               float* __restrict__ amp) {
  const int lane = threadIdx.x;
  const int n0 = blockIdx.x * 16;
  const int m0 = blockIdx.y * 16;
  const int m = lane & 15;
  const int half = lane >> 4;
  const int col = n0 + (lane & 15);
  const float* __restrict__ wa = w2;                       // w2[0]: [1024][256]
  const float* __restrict__ wb = w2 + HIDDEN * NUM_FREQS;  // w2[1]

  v8f accRA = {}, accRB = {}, accIA = {}, accIB = {};
  for (int k0 = 0; k0 < HIDDEN; k0 += 4) {
    const int ka = k0 + 2 * half;
    v2f aR, aI, bA, bB;
    const int arow = (m0 + m) * HIDDEN + ka;
    aR.x = o1r[arow];    aR.y = o1r[arow + 1];
    aI.x = o1i[arow];    aI.y = o1i[arow + 1];
    bA.x = wa[(long)ka * NUM_FREQS + col];
    bA.y = wa[(long)(ka + 1) * NUM_FREQS + col];
    bB.x = wb[(long)ka * NUM_FREQS + col];
    bB.y = wb[(long)(ka + 1) * NUM_FREQS + col];

    accRA = __builtin_amdgcn_wmma_f32_16x16x4_f32(false, aR, false, bA, (short)0, accRA, false, false);
    accRB = __builtin_amdgcn_wmma_f32_16x16x4_f32(false, aR, false, bB, (short)0, accRB, false, false);
    accIA = __builtin_amdgcn_wmma_f32_16x16x4_f32(false, aI, false, bA, (short)0, accIA, false, false);
    accIB = __builtin_amdgcn_wmma_f32_16x16x4_f32(false, aI, false, bB, (short)0, accIB, false, false);
  }
  const float br = b2[col];
  const float bi = b2[NUM_FREQS + col];
#pragma unroll
  for (int v = 0; v < 8; ++v) {
    const int row = m0 + v + 8 * half;
    const float o2r = accRA[v] - accIB[v] + br;
    const float o2i = accIA[v] + accRB[v] + bi;
    amp[row * NUM_FREQS + col] = sqrtf(o2r * o2r + o2i * o2i);
  }
}

// ---------------------------------------------------------------------------
// Kernel 5: logits = amp @ w_gate; top-3 -> softmax -> scatter into gates.
// One block per batch row. Serial strict-'>' top-k matches JAX tie-breaking
// (lowest index wins), so all-zero logits -> experts {0,1,2} at 1/3 each.
// ---------------------------------------------------------------------------
__global__ __launch_bounds__(64) void k_gate(const float* __restrict__ amp,
                                             const float* __restrict__ w_gate,
                                             float* __restrict__ out) {
  __shared__ float a[NUM_FREQS];
  __shared__ float logits[NUM_EXPERTS];
  __shared__ float gates[NUM_EXPERTS];
  const int b = blockIdx.x;
  const int tid = threadIdx.x;
  for (int i = tid; i < NUM_FREQS; i += 64) a[i] = amp[b * NUM_FREQS + i];
  __syncthreads();
  if (tid < NUM_EXPERTS) {
    float acc = 0.f;
    for (int f = 0; f < NUM_FREQS; ++f)
      acc = fmaf(a[f], w_gate[f * NUM_EXPERTS + tid], acc);
    logits[tid] = acc;
  }
  __syncthreads();
  if (tid == 0) {
    bool used[NUM_EXPERTS];
    int idx[TOPK];
    float val[TOPK];
    for (int e = 0; e < NUM_EXPERTS; ++e) { used[e] = false; gates[e] = 0.f; }
    for (int t = 0; t < TOPK; ++t) {
      int best = 0; float bv = -3.402823466e38f;
      for (int e = 0; e < NUM_EXPERTS; ++e)
        if (!used[e] && logits[e] > bv) { bv = logits[e]; best = e; }
      used[best] = true; idx[t] = best; val[t] = bv;
    }
    float mx = val[0];
    for (int t = 1; t < TOPK; ++t) mx = fmaxf(mx, val[t]);
    float ex[TOPK]; float sum = 0.f;
    for (int t = 0; t < TOPK; ++t) { ex[t] = __expf(val[t] - mx); sum += ex[t]; }
    for (int t = 0; t < TOPK; ++t) gates[idx[t]] = ex[t] / sum;
  }
  __syncthreads();
  if (tid < NUM_EXPERTS) out[b * NUM_EXPERTS + tid] = gates[tid];
}

// ---------------------------------------------------------------------------
extern "C" void kernel_launch(void* const* d_in, const int* in_sizes, int n_in,
                              void* d_out, int out_size, void* d_ws, size_t ws_size,
                              hipStream_t stream) {
  (void)in_sizes; (void)n_in; (void)out_size; (void)ws_size;
  const float* x      = (const float*)d_in[0];
  const float* fc_w   = (const float*)d_in[1];
  const float* fc_b   = (const float*)d_in[2];
  const float* w1     = (const float*)d_in[3];
  const float* b1     = (const float*)d_in[4];
  const float* w2     = (const float*)d_in[5];
  const float* b2     = (const float*)d_in[6];
  const float* w_gate = (const float*)d_in[7];
  float* out = (float*)d_out;

  float* ws  = (float*)d_ws;
  float* s   = ws;                              // 32*512
  float* re  = s   + BATCH * SEQ_LEN;           // 32*256
  float* im  = re  + BATCH * NUM_FREQS;         // 32*256
  float* o1r = im  + BATCH * NUM_FREQS;         // 32*1024
  float* o1i = o1r + BATCH * HIDDEN;            // 32*1024
  float* amp = o1i + BATCH * HIDDEN;            // 32*256

  k_reduce<<<BATCH * SEQ_LEN, 256, 0, stream>>>(x, fc_w, fc_b, s);
  k_rfft<<<BATCH, 256, 0, stream>>>(s, re, im);
  k_layer1<<<dim3(HIDDEN / 16, BATCH / 16), 32, 0, stream>>>(re, im, w1, b1, o1r, o1i);
  k_layer2<<<dim3(NUM_FREQS / 16, BATCH / 16), 32, 0, stream>>>(o1r, o1i, w2, b2, amp);
  k_gate<<<BATCH, 64, 0, stream>>>(amp, w_gate, out);
}